// LlamaAttention_40879498729201
// MI455X (gfx1250) — compile-verified
//
#include <hip/hip_runtime.h>
#include <math.h>

// ---- problem constants (from reference) ----
#define HIDDEN 4096
#define NH     32
#define NKV    8
#define HD     128
#define GQ     (NH / NKV)            // 4 q-heads per kv-head
#define LCTX   4096
#define BSZ    16                    // paged-cache block size
#define BB     16                    // batch
#define MAXB   385
#define QKV_N  ((NH + 2 * NKV) * HD) // 6144

typedef __attribute__((ext_vector_type(2))) float v2f;
typedef __attribute__((ext_vector_type(8))) float v8f;
typedef int v4i __attribute__((vector_size(4 * sizeof(int))));

__device__ __forceinline__ float inv_freq_of(int j) {
  // theta^(-2j/HD), theta = 10000 ; ln(10000) = 9.210340371976184
  return __expf(-9.210340371976184f * (2.0f * (float)j) / (float)HD);
}

// ---------------------------------------------------------------------------
// Async global->LDS helpers (gfx1250 GLOBAL_LOAD_ASYNC_TO_LDS_B128 path,
// ASYNCcnt-tracked). Guarded: falls back to a synchronous 16B copy.
// Builtin signature (per hipcc diagnostic): params are v4i* in AS1 / AS3.
// ---------------------------------------------------------------------------
#if __has_builtin(__builtin_amdgcn_global_load_async_to_lds_b128)
#define ASYNC_LDS 1
#else
#define ASYNC_LDS 0
#endif

__device__ __forceinline__ void async_copy16(const float* src, float* lds_dst) {
#if ASYNC_LDS
  __builtin_amdgcn_global_load_async_to_lds_b128(
      (__attribute__((address_space(1))) v4i*)(void*)src,
      (__attribute__((address_space(3))) v4i*)(void*)lds_dst,
      0, 0);
#else
  const float4 v = *(const float4*)src;
  *(float4*)lds_dst = v;
#endif
}

__device__ __forceinline__ void async_join() {
#if ASYNC_LDS
#if __has_builtin(__builtin_amdgcn_s_wait_asynccnt)
  __builtin_amdgcn_s_wait_asynccnt(0);
#else
  asm volatile("s_wait_asynccnt 0x0" ::: "memory");
#endif
#endif
  __syncthreads();
}

// ---------------------------------------------------------------------------
// Kernel 1/6: C[16,N] = A[16,K] @ B[K,N], fp32 WMMA 16x16x4, one tile / wave.
// A-layout (ISA 7.12.2, 32-bit A 16x4): lane m = lane&15, VGPR j holds
// K = k0 + 2*(lane>>4) + j.  B/C row-striped across lanes within a VGPR.
// ---------------------------------------------------------------------------
__global__ void wmma_gemm16(const float* __restrict__ A,
                            const float* __restrict__ Bm,
                            float* __restrict__ C, int K, int N) {
  const int lane = threadIdx.x;
  const int hi = lane >> 4;
  const int lo = lane & 15;
  const int n0 = blockIdx.x * 16;

  v8f acc = {0.f, 0.f, 0.f, 0.f, 0.f, 0.f, 0.f, 0.f};
  const float* arow = A + (size_t)lo * K;
  const float* bcol = Bm + n0 + lo;

#pragma unroll 4
  for (int k0 = 0; k0 < K; k0 += 4) {
    const int kk = k0 + 2 * hi;
    __builtin_prefetch(bcol + (size_t)(kk + 16) * N, 0, 1);
    v2f a;
    a[0] = arow[kk];
    a[1] = arow[kk + 1];
    v2f b;
    b[0] = bcol[(size_t)kk * N];
    b[1] = bcol[(size_t)(kk + 1) * N];
    acc = __builtin_amdgcn_wmma_f32_16x16x4_f32(false, a, false, b,
                                                (short)0, acc, false, false);
  }

  float* crow = C + n0 + lo;
#pragma unroll
  for (int r = 0; r < 8; ++r) {
    const int m = (hi ? 8 : 0) + r;
    crow[(size_t)m * N] = acc[r];
  }
}

// ---------------------------------------------------------------------------
// Kernel 2/6: neox RoPE on freshly projected q (32 heads) and k (8 heads).
// ---------------------------------------------------------------------------
__global__ void rope_qk(float* __restrict__ qkv, const int* __restrict__ positions) {
  const int b = blockIdx.y;
  const int h = blockIdx.x;           // 0..NH+NKV-1
  const int j = threadIdx.x;          // 0..63
  int pos = positions[b];
  if (pos > LCTX - 1) pos = LCTX - 1;
  float* ptr = qkv + (size_t)b * QKV_N +
               (h < NH ? h * HD : NH * HD + (h - NH) * HD);
  const float ang = (float)pos * inv_freq_of(j);
  const float cs = __cosf(ang), sn = __sinf(ang);
  const float x1 = ptr[j], x2 = ptr[j + 64];
  ptr[j]      = x1 * cs - x2 * sn;
  ptr[j + 64] = x2 * cs + x1 * sn;
}

// ---------------------------------------------------------------------------
// Kernel 3/6: scores tile. grid = (L/16, NKV, B), 32 threads (one wave).
// Every 16-position tile maps to ONE physical cache block, and the cache
// layout [phys][kv][HD][BS] makes the K tile (128x16 f32 = 8KB) contiguous
// in memory-order [dim][pos] (= K^T). Stage it with async B128 copies, RoPE
// it into [pos][dim], then scores = Q.K^T as 32 f32 WMMAs.
// ---------------------------------------------------------------------------
__global__ void attn_scores(const float* __restrict__ qkv,
                            const float* __restrict__ key_cache,
                            const int* __restrict__ block_tables,
                            const int* __restrict__ ctx_ptr,
                            float* __restrict__ scores) {
  __shared__ float Kt[HD][16];     // raw, memory order [dim][pos]
  __shared__ float Kr[16][HD];     // roped, [pos][dim]
  __shared__ float Q_lds[16][HD];

  const int lane = threadIdx.x;
  const int hi = lane >> 4, lo = lane & 15;
  const int tile = blockIdx.x;     // 0..255
  const int kv = blockIdx.y;
  const int b = blockIdx.z;

  const int ctx = *ctx_ptr;
  const int num_past = ctx - 1;
  const int bti = (num_past - LCTX) / BSZ + 1;
  const int startp = num_past - LCTX + 1 + BSZ;

  // tile-uniform paging: all 16 positions share one logical/physical block
  const int logical = (tile == 0) ? 0 : (bti + tile);
  const int phys = block_tables[b * MAXB + logical];
  const float* kcbase = key_cache + ((size_t)phys * NKV + kv) * HD * BSZ;

  // stage the contiguous 8KB K tile: 16 x B128 per lane, ASYNCcnt-tracked
#pragma unroll
  for (int ii = 0; ii < 16; ++ii) {
    const int f = (lane + ii * 32) * 4;          // float index in tile
    async_copy16(kcbase + f, &Kt[0][0] + f);
  }

  // Q tile while the copy is in flight: rows 0..3 = heads kv*4+r, rest zero
  const float* qbase = qkv + (size_t)b * QKV_N + (size_t)(kv * GQ) * HD;
  for (int i = lane; i < 16 * HD; i += 32) {
    const int row = i >> 7, col = i & 127;
    Q_lds[row][col] = (row < GQ) ? qbase[row * HD + col] : 0.0f;
  }

  async_join();

  // per-lane position lo of this tile
  const int abs_pos = logical * BSZ + lo;
  const bool cur = (abs_pos == num_past);
  const bool rot = (abs_pos >= startp) && (abs_pos < num_past);
  const float* knew = qkv + (size_t)b * QKV_N + (size_t)NH * HD + (size_t)kv * HD;

  const int j0 = hi * 32;  // lanes 0-15: dims 0..31(+64); lanes 16-31: 32..63(+64)
  for (int jj = 0; jj < 32; ++jj) {
    const int j = j0 + jj;
    float x1 = Kt[j][lo];
    float x2 = Kt[j + 64][lo];
    if (cur) {
      x1 = knew[j];
      x2 = knew[j + 64];
      Kr[lo][j] = x1;
      Kr[lo][j + 64] = x2;
    } else if (rot) {
      const float ang = (float)(abs_pos - startp) * inv_freq_of(j);
      const float cs = __cosf(ang), sn = __sinf(ang);
      Kr[lo][j]      = x1 * cs - x2 * sn;
      Kr[lo][j + 64] = x2 * cs + x1 * sn;
    } else {
      Kr[lo][j] = x1;
      Kr[lo][j + 64] = x2;
    }
  }
  __syncthreads();

  v8f acc = {0.f, 0.f, 0.f, 0.f, 0.f, 0.f, 0.f, 0.f};
#pragma unroll 8
  for (int k0 = 0; k0 < HD; k0 += 4) {
    const int kk = k0 + 2 * hi;
    v2f a;
    a[0] = Q_lds[lo][kk];
    a[1] = Q_lds[lo][kk + 1];
    v2f bv;                        // B[k][n] = K[n][k], n = lo
    bv[0] = Kr[lo][kk];
    bv[1] = Kr[lo][kk + 1];
    acc = __builtin_amdgcn_wmma_f32_16x16x4_f32(false, a, false, bv,
                                                (short)0, acc, false, false);
  }

  const float scale = 0.08838834764831845f;   // 128^-0.5
  const bool validn = (abs_pos <= num_past);  // per position n = lo
  if (hi == 0) {  // lanes 0-15 hold rows m = 0..7 in VGPRs 0..7; rows 0..3 valid
    float* srow = scores + (size_t)((b * NKV + kv) * GQ) * LCTX + tile * 16 + lo;
#pragma unroll
    for (int r = 0; r < GQ; ++r)
      srow[(size_t)r * LCTX] = validn ? acc[r] * scale : -3.0e38f;
  }
}

// ---------------------------------------------------------------------------
// Kernel 4/6: row softmax over L=4096. grid = B*NKV*GQ rows, 256 threads.
// ---------------------------------------------------------------------------
__global__ void softmax_rows(float* __restrict__ scores) {
  __shared__ float red[256];
  float* s = scores + (size_t)blockIdx.x * LCTX;
  const int tid = threadIdx.x;

  float m = -3.4e38f;
  for (int i = tid; i < LCTX; i += 256) m = fmaxf(m, s[i]);
  red[tid] = m;
  __syncthreads();
  for (int st = 128; st > 0; st >>= 1) {
    if (tid < st) red[tid] = fmaxf(red[tid], red[tid + st]);
    __syncthreads();
  }
  m = red[0];
  __syncthreads();

  float sum = 0.f;
  for (int i = tid; i < LCTX; i += 256) {
    const float e = __expf(s[i] - m);
    s[i] = e;
    sum += e;
  }
  red[tid] = sum;
  __syncthreads();
  for (int st = 128; st > 0; st >>= 1) {
    if (tid < st) red[tid] += red[tid + st];
    __syncthreads();
  }
  const float inv = 1.0f / red[0];
  for (int i = tid; i < LCTX; i += 256) s[i] *= inv;
}

// ---------------------------------------------------------------------------
// Kernel 5/6: attn[4,HD] += P[4,L] @ V_gather[L,HD].
// grid = (HD/16, NKV, B): 1024 waves. Per tile the needed V sub-block
// (dims nt*16..+16, all 16 positions) is 1KB contiguous -> 2 async B128
// copies per lane; current-token V substituted at read time.
// ---------------------------------------------------------------------------
__global__ void attn_pv(const float* __restrict__ qkv,
                        const float* __restrict__ value_cache,
                        const int* __restrict__ block_tables,
                        const int* __restrict__ ctx_ptr,
                        const float* __restrict__ P,
                        float* __restrict__ attn) {
  __shared__ float Vt[16][16];   // memory order [dim_local][pos]
  __shared__ float P_lds[16][16];

  const int lane = threadIdx.x;
  const int hi = lane >> 4, lo = lane & 15;
  const int nt = blockIdx.x;     // dim tile 0..7
  const int kv = blockIdx.y;
  const int b = blockIdx.z;

  const int ctx = *ctx_ptr;
  const int num_past = ctx - 1;
  const int bti = (num_past - LCTX) / BSZ + 1;

  const float* vnew = qkv + (size_t)b * QKV_N + (size_t)(NH + NKV) * HD + (size_t)kv * HD;
  const float* pbase0 = P + (size_t)((b * NKV + kv) * GQ) * LCTX;
  const float vsub = vnew[nt * 16 + lo];  // this lane's dim of the current token V

  v8f acc = {0.f, 0.f, 0.f, 0.f, 0.f, 0.f, 0.f, 0.f};

  for (int tile = 0; tile < LCTX / 16; ++tile) {
    const int logical = (tile == 0) ? 0 : (bti + tile);
    const int phys = block_tables[b * MAXB + logical];
    const int off_cur = num_past - logical * BSZ;  // in [0,16) iff tile holds cur
    const float* vcbase = value_cache + ((size_t)phys * NKV + kv) * HD * BSZ
                        + (size_t)nt * 16 * BSZ;   // 1KB contiguous [16dim][16pos]

    async_copy16(vcbase + lane * 4, &Vt[0][0] + lane * 4);
    async_copy16(vcbase + 128 + lane * 4, &Vt[0][0] + 128 + lane * 4);

    // P tile: rows 0..3 real, rows 4..15 zero
#pragma unroll
    for (int ii = 0; ii < 8; ++ii) {
      const int i = lane * 8 + ii;
      const int row = i >> 4, col = i & 15;
      P_lds[row][col] = (row < GQ) ? pbase0[(size_t)row * LCTX + tile * 16 + col]
                                   : 0.0f;
    }

    async_join();

#pragma unroll
    for (int k0 = 0; k0 < 16; k0 += 4) {
      const int kk = k0 + 2 * hi;
      v2f a;                        // A = P[m][k], m = lo
      a[0] = P_lds[lo][kk];
      a[1] = P_lds[lo][kk + 1];
      v2f bv;                       // B = V[k=pos][n=dim], n = lo
      bv[0] = (kk == off_cur)     ? vsub : Vt[lo][kk];
      bv[1] = (kk + 1 == off_cur) ? vsub : Vt[lo][kk + 1];
      acc = __builtin_amdgcn_wmma_f32_16x16x4_f32(false, a, false, bv,
                                                  (short)0, acc, false, false);
    }
    __syncthreads();   // protect Vt/P_lds before next tile's copies
  }

  if (hi == 0) {   // rows 0..3 live in VGPRs 0..3 of lanes 0..15
    float* arow = attn + (size_t)b * (NH * HD) + (size_t)(kv * GQ) * HD;
#pragma unroll
    for (int r = 0; r < GQ; ++r)
      arow[(size_t)r * HD + nt * 16 + lo] = acc[r];
  }
}

// ---------------------------------------------------------------------------
// launch
// ---------------------------------------------------------------------------
extern "C" void kernel_launch(void* const* d_in, const int* in_sizes, int n_in,
                              void* d_out, int out_size, void* d_ws, size_t ws_size,
                              hipStream_t stream) {
  const int*   positions    = (const int*)d_in[0];
  const float* hidden       = (const float*)d_in[1];
  const float* key_cache    = (const float*)d_in[2];
  const float* value_cache  = (const float*)d_in[3];
  const int*   block_tables = (const int*)d_in[4];
  const int*   ctx          = (const int*)d_in[5];
  const float* w_qkv        = (const float*)d_in[6];
  const float* w_o          = (const float*)d_in[7];
  float* out = (float*)d_out;

  char* ws = (char*)d_ws;
  float* qkv    = (float*)ws;                                  // 16*6144 f32
  float* scores = (float*)(ws + (size_t)BB * QKV_N * 4);       // 512*4096 f32
  float* attnb  = (float*)(ws + (size_t)BB * QKV_N * 4
                              + (size_t)BB * NKV * GQ * LCTX * 4); // 16*4096 f32

  // 1) QKV projection: [16,4096] x [4096,6144]
  wmma_gemm16<<<dim3(QKV_N / 16), 32, 0, stream>>>(hidden, w_qkv, qkv, HIDDEN, QKV_N);
  // 2) RoPE on fresh q/k
  rope_qk<<<dim3(NH + NKV, BB), 64, 0, stream>>>(qkv, positions);
  // 3) scores = Q K^T * scale (masked)
  attn_scores<<<dim3(LCTX / 16, NKV, BB), 32, 0, stream>>>(qkv, key_cache,
                                                           block_tables, ctx, scores);
  // 4) softmax over L
  softmax_rows<<<dim3(BB * NKV * GQ), 256, 0, stream>>>(scores);
  // 5) attn = P V
  attn_pv<<<dim3(HD / 16, NKV, BB), 32, 0, stream>>>(qkv, value_cache, block_tables,
                                                     ctx, scores, attnb);
  // 6) output projection: [16,4096] x [4096,4096]
  wmma_gemm16<<<dim3(HIDDEN / 16), 32, 0, stream>>>(attnb, w_o, out, NH * HD, HIDDEN);
}